// Spherization_81277961110016
// MI455X (gfx1250) — compile-verified
//
#include <hip/hip_runtime.h>
#include <hip/hip_bf16.h>
#include <math.h>

// Spherization forward for MI455X (gfx1250, wave32).
//
// Per row of 256 features:
//   ang_j = (pi/2 - phi_L) * sigmoid(s * x_j) + phi_L
//   ls_j  = log2(sin(ang_j) + 1e-6),  lc_j = log2(cos(ang_j) + 1e-6)
//   out[k]   = r * exp2( sum_{j<k} ls_j + lc_k ),  k = 0..255
//   out[256] = r * exp2( sum_{j} ls_j )
// (log2/exp2 space == reference's ln/exp space: exp2(sum log2 t) = prod t.)
//
// Memory floor: 1.03 GB traffic -> ~44us @ 23.3 TB/s. To keep VALU below
// that floor: sigmoid via v_tanh_f32 (1 trans instead of exp+rcp), native
// v_log_f32 / v_exp_f32 (base-2, no conversion muls). The 256-wide exclusive
// prefix sum rides the matrix pipe: row as 16x16 A, P = A * S (strictly-upper
// ones) via 4 chained V_WMMA_F32_16X16X4_F32, co-executing with VALU/trans.

typedef __attribute__((ext_vector_type(2))) float v2f;
typedef __attribute__((ext_vector_type(8))) float v8f;

#define NDIM 256
#define WAVES_PER_BLOCK 8
#define ROW_PAD 17              // 16 cols padded to 17 -> conflict-free LDS rows
#define PI_F 3.14159265358979323846f
#define BOUND_F 1e-6f

__device__ __forceinline__ float fast_exp2(float z) {
    return __builtin_amdgcn_exp2f(z);       // v_exp_f32 (base-2)
}

__device__ __forceinline__ float fast_tanh(float z) {
#if __has_builtin(__builtin_amdgcn_tanhf)
    return __builtin_amdgcn_tanhf(z);       // v_tanh_f32 (gfx1250 trans op)
#else
    // tanh(z) = 1 - 2/(exp2(2*log2e*z) + 1)
    float e = fast_exp2(2.885390082f * z);
    return 1.0f - 2.0f / (e + 1.0f);
#endif
}

__global__ __launch_bounds__(256) void spherization_kernel(
    const float* __restrict__ x,
    const float* __restrict__ phiL_p,
    const float* __restrict__ radius_p,
    const float* __restrict__ scaling_p,
    float* __restrict__ out,
    int n_rows)
{
    __shared__ float lds_ls[WAVES_PER_BLOCK][16 * ROW_PAD];
    __shared__ float lds_lc[WAVES_PER_BLOCK][16 * ROW_PAD];
    __shared__ float lds_tot[WAVES_PER_BLOCK][16];

    const int tid  = threadIdx.x;
    const int wave = tid >> 5;
    const int lane = tid & 31;

    int row = blockIdx.x * WAVES_PER_BLOCK + wave;
    if (row >= n_rows) row = n_rows - 1;   // keep all waves alive (syncthreads)

    const float phiL    = phiL_p[0];
    const float radius  = radius_p[0];
    const float scaling = scaling_p[0];
    const float amp     = (PI_F * 0.5f) - phiL;
    // ang = amp*(0.5*tanh(0.5*s*x)+0.5) + phiL = c1*tanh(c2*x) + c0
    const float c2 = 0.5f * scaling;
    const float c1 = 0.5f * amp;
    const float c0 = phiL + 0.5f * amp;

    // ---- 1) coalesced load: lane takes 8 contiguous floats of its row ----
    const float*  xr = x + (size_t)row * NDIM;
    const float4* x4 = (const float4*)xr;
    float4 xa = x4[lane * 2 + 0];
    float4 xb = x4[lane * 2 + 1];
    float xs[8] = {xa.x, xa.y, xa.z, xa.w, xb.x, xb.y, xb.z, xb.w};

    float lsv[8], lcv[8];
#pragma unroll
    for (int e = 0; e < 8; ++e) {
        float ang = __builtin_fmaf(c1, fast_tanh(c2 * xs[e]), c0);
        lsv[e] = __log2f(__sinf(ang) + BOUND_F);
        lcv[e] = __log2f(__cosf(ang) + BOUND_F);
    }

    // elements k = 8*lane + e live in matrix row i0 = lane>>1, cols j0..j0+7
    const int i0 = lane >> 1;
    const int j0 = (lane & 1) * 8;
    float* lsp = &lds_ls[wave][i0 * ROW_PAD + j0];
    float* lcp = &lds_lc[wave][i0 * ROW_PAD + j0];
#pragma unroll
    for (int e = 0; e < 8; ++e) { lsp[e] = lsv[e]; lcp[e] = lcv[e]; }

    __syncthreads();

    // ---- 2) P = A * S via 4 chained wmma_f32_16x16x4_f32 (K=16 in 4x4) ----
    // A-operand (16x4 f32): lanes 0-15 hold M=lane, K = {4k,4k+1};
    //                       lanes 16-31 hold M=lane-16, K = {4k+2,4k+3}.
    // B-operand mirrors that K split; S[k][n] = (k < n) built in registers.
    const int mi = lane & 15;   // A row (lo half) / B column index n
    const int hi = lane >> 4;

    v8f acc = {};
#pragma unroll
    for (int kk = 0; kk < 4; ++kk) {
        const int kbase = 4 * kk + 2 * hi;
        v2f a, b;
        a.x = lds_ls[wave][mi * ROW_PAD + kbase + 0];
        a.y = lds_ls[wave][mi * ROW_PAD + kbase + 1];
        b.x = (kbase + 0 < mi) ? 1.0f : 0.0f;   // strictly-upper S
        b.y = (kbase + 1 < mi) ? 1.0f : 0.0f;
        acc = __builtin_amdgcn_wmma_f32_16x16x4_f32(
                  false, a, false, b, (short)0, acc, false, false);
    }
    // acc holds P in C/D layout: vgpr v, lane l -> (M = v + 8*hi, N = mi)
    // P[m][n] = sum_{j<n} ls[16m + j]   (within-segment exclusive prefix)

    // ---- 3) segment totals: tot[m] = P[m][15] + ls[16m+15] ----
    if (mi == 15) {
#pragma unroll
        for (int v = 0; v < 8; ++v) {
            int m = v + 8 * hi;
            lds_tot[wave][m] = acc[v] + lds_ls[wave][m * ROW_PAD + 15];
        }
    }
    __syncthreads();

    // ---- 4) 16-wide exclusive scan of segment totals (in registers) ----
    float pref[16];
    float run = 0.0f;
#pragma unroll
    for (int i = 0; i < 16; ++i) { pref[i] = run; run += lds_tot[wave][i]; }
    // run == grand total of all 256 ls values (log2 space)

    // ---- 5) outputs ----
    float* outr = out + (size_t)row * (NDIM + 1);
#pragma unroll
    for (int v = 0; v < 8; ++v) {
        int m = v + 8 * hi;
        float E  = acc[v] + pref[m];
        float lc = lds_lc[wave][m * ROW_PAD + mi];
        outr[16 * m + mi] = radius * fast_exp2(E + lc);
    }
    if (lane == 0) {
        outr[NDIM] = radius * fast_exp2(run);  // duplicated-angle column
    }
}

extern "C" void kernel_launch(void* const* d_in, const int* in_sizes, int n_in,
                              void* d_out, int out_size, void* d_ws, size_t ws_size,
                              hipStream_t stream) {
    // setup_inputs order: x, W_theta, W_phi, b_phi, phi_L, radius, scaling
    const float* x       = (const float*)d_in[0];
    const float* phiL    = (const float*)d_in[4];
    const float* radius  = (const float*)d_in[5];
    const float* scaling = (const float*)d_in[6];
    float* out = (float*)d_out;

    const int n_rows = in_sizes[0] / NDIM;
    const int blocks = (n_rows + WAVES_PER_BLOCK - 1) / WAVES_PER_BLOCK;
    spherization_kernel<<<blocks, 256, 0, stream>>>(x, phiL, radius, scaling,
                                                    out, n_rows);
}